// SSDBoxRPNHead_3100966388136
// MI455X (gfx1250) — compile-verified
//
#include <hip/hip_runtime.h>
#include <hip/hip_bf16.h>
#include <stdint.h>

// Problem constants (match reference)
#define Bc 32
#define Nc 32768
#define Cc 21
#define Kc 100
#define NCLS 20           // C-1 foreground classes
#define CENTER_VAR 0.1f
#define SIZE_VAR 0.2f
#define CONF_T 0.05f
#define IOU_T 0.45f

typedef unsigned int v4u __attribute__((ext_vector_type(4)));
typedef int          v8i __attribute__((ext_vector_type(8)));
typedef int          v4i __attribute__((ext_vector_type(4)));

// -------------------------------------------------------------------------
// TDM: 1-D tile load (nelem f32, contiguous) global -> LDS via the Tensor
// Data Mover.  Descriptor per CDNA5 ISA.
// Issued by a single wave; caller must s_wait_tensorcnt + barrier.
// Toolchain here is the clang-23 lane: 6-arg builtin
//   (uint32x4 g0, int32x8 g1, int32x4 g2, int32x4 g3, int32x8 pad, i32 cpol)
// -------------------------------------------------------------------------
__device__ __forceinline__ void tdm_load_f32_1d(unsigned lds_off,
                                                const void* gptr,
                                                unsigned nelem /* fits tile_dim0 (16b) */) {
  unsigned long long ga = (unsigned long long)(uintptr_t)gptr;
  // Group 0: flags / lds_addr / global_addr / type
  v4u g0 = {
    1u,                                              // count=1, is_restore=0, gather=0
    lds_off,                                         // lds_addr (bytes)
    (unsigned)(ga & 0xFFFFFFFFull),                  // global_addr[31:0]
    (unsigned)((ga >> 32) & 0x1FFFFFFull) | 0x80000000u  // ga[56:32], type=2
  };
  // Group 1: dims.  data_size=2 (4B).  tensor_dim0 = tile_dim0 = nelem,
  // tensor_dim1 = tile_dim1 = 1, strides = nelem.
  v8i g1 = {
    0x00020000,                                      // wg_mask=0, data_size=4B
    (int)((nelem & 0xFFFFu) << 16),                  // abar=0 | tensor_dim0[15:0]
    (int)(((nelem >> 16) & 0xFFFFu) | (1u << 16)),   // tensor_dim0[31:16] | tensor_dim1.lo=1
    (int)((nelem & 0xFFFFu) << 16),                  // tensor_dim1.hi=0 | tile_dim0
    1,                                               // tile_dim1=1 | tile_dim2=0
    (int)nelem,                                      // tensor_dim0_stride[31:0]
    (int)((nelem & 0xFFFFu) << 16),                  // stride0[47:32]=0 | stride1[15:0]
    (int)((nelem >> 16) & 0xFFFFu)                   // stride1[47:16]
  };
  v4i z4 = {0, 0, 0, 0};
  v8i z8 = {0, 0, 0, 0, 0, 0, 0, 0};
  __builtin_amdgcn_tensor_load_to_lds(g0, g1, z4, z4, z8, 0);
}

// -------------------------------------------------------------------------
// Kernel 1: softmax over C=21, write classes 1..20 transposed to scratch
// layout scores[(b*20 + c-1) * N + n]  (contiguous for the top-k pass).
// One thread per anchor row; rows staged through LDS by the TDM.
// -------------------------------------------------------------------------
__global__ void __launch_bounds__(256)
softmax_transpose_kernel(const float* __restrict__ logits,
                         float* __restrict__ scores) {
  __shared__ float tile[256 * Cc];   // 21504 B, contiguous block of 256 rows
  const int t = threadIdx.x;
  const long long rowBase = (long long)blockIdx.x * 256;

  if (t == 0) {
    tdm_load_f32_1d((unsigned)(uintptr_t)&tile[0],
                    logits + rowBase * Cc, 256 * Cc);
    __builtin_amdgcn_s_wait_tensorcnt(0);
  }
  __syncthreads();

  float l[Cc];
  float m = -1e30f;
#pragma unroll
  for (int c = 0; c < Cc; ++c) {
    l[c] = tile[t * Cc + c];          // stride 21 (odd) -> conflict-free
    m = fmaxf(m, l[c]);
  }
  float s = 0.0f;
#pragma unroll
  for (int c = 0; c < Cc; ++c) {
    l[c] = expf(l[c] - m);
    s += l[c];
  }
  const float inv = 1.0f / s;

  const long long r = rowBase + t;
  const int b = (int)(r >> 15);       // N = 2^15
  const int n = (int)(r & (Nc - 1));
#pragma unroll
  for (int c = 1; c < Cc; ++c) {
    scores[((long long)(b * NCLS + (c - 1)) << 15) + n] = l[c] * inv;
  }
}

// -------------------------------------------------------------------------
// Kernel 2: exact top-100 of 32768 scores per (b, class).
// Whole column TDM-loaded into LDS (128 KB); 512 segment maxima;
// 100 tournament extractions with lower-index tie-break (== lax.top_k).
// Dynamic LDS: 32768 + 4*512 floats = 139264 B.
// -------------------------------------------------------------------------
__global__ void __launch_bounds__(512)
topk_kernel(const float* __restrict__ scores,
            float* __restrict__ topv,
            int* __restrict__ topi) {
  extern __shared__ float sm[];
  float* sc   = sm;                        // 32768
  float* segv = sm + 32768;                // 512
  int*   segi = (int*)(sm + 33280);        // 512
  float* rv   = sm + 33792;                // 512
  int*   ri   = (int*)(sm + 34304);        // 512

  const int t  = threadIdx.x;
  const int bc = blockIdx.x;               // b*20 + c'

  if (t == 0) {
    tdm_load_f32_1d((unsigned)(uintptr_t)sc,
                    scores + (long long)bc * Nc, Nc);
    __builtin_amdgcn_s_wait_tensorcnt(0);
  }
  __syncthreads();

  // Per-segment maxima (64 elems each, scan ascending, strict '>' keeps
  // the lowest index on ties).
  {
    const int base = t << 6;
    float mv = -1.0f; int mi = base;
    for (int j = 0; j < 64; ++j) {
      const float v = sc[base + j];
      if (v > mv) { mv = v; mi = base + j; }
    }
    segv[t] = mv; segi[t] = mi;
  }
  __syncthreads();

  for (int k = 0; k < Kc; ++k) {
    rv[t] = segv[t]; ri[t] = segi[t];
    __syncthreads();
    for (int s = 256; s > 0; s >>= 1) {
      if (t < s) {
        const float a = rv[t],     b = rv[t + s];
        const int   ia = ri[t],    ib = ri[t + s];
        if (b > a || (b == a && ib < ia)) { rv[t] = b; ri[t] = ib; }
      }
      __syncthreads();
    }
    const int   win = ri[0];
    if (t == 0) {
      topv[bc * Kc + k] = rv[0];
      topi[bc * Kc + k] = win;
    }
    const int wseg = win >> 6;
    if (t == wseg) {                       // invalidate + rescan one segment
      sc[win] = -1.0f;                     // softmax scores > 0
      const int base = wseg << 6;
      float mv = -1.0f; int mi = base;
      for (int j = 0; j < 64; ++j) {
        const float v = sc[base + j];
        if (v > mv) { mv = v; mi = base + j; }
      }
      segv[t] = mv; segi[t] = mi;
    }
    __syncthreads();
  }
}

// -------------------------------------------------------------------------
// Kernel 3: gather top-100 candidates, decode boxes, pairwise-IoU NMS
// (exact replica of the reference fori_loop), write all three outputs.
// -------------------------------------------------------------------------
__global__ void __launch_bounds__(128)
nms_kernel(const float* __restrict__ bbox_pred,
           const float* __restrict__ priors,
           const float* __restrict__ topv,
           const int* __restrict__ topi,
           float* __restrict__ out_boxes,   // [B*20*K*4]
           float* __restrict__ out_scores,  // [B*20*K]
           float* __restrict__ out_keep) {  // [B*20*K]
  __shared__ float bx0[Kc], by0[Kc], bx1[Kc], by1[Kc], area[Kc], vals[Kc];
  __shared__ int keepS[Kc];

  const int bc = blockIdx.x;
  const int b  = bc / NCLS;
  const int k  = threadIdx.x;

  if (k < Kc) {
    const int   i = topi[bc * Kc + k];
    const float v = topv[bc * Kc + k];
    const float* bp = bbox_pred + ((long long)b * Nc + i) * 4;
    const float* pr = priors + (long long)i * 4;
    const float pcx = pr[0], pcy = pr[1], pw = pr[2], ph = pr[3];
    const float cx = bp[0] * CENTER_VAR * pw + pcx;
    const float cy = bp[1] * CENTER_VAR * ph + pcy;
    const float w  = expf(bp[2] * SIZE_VAR) * pw;
    const float h  = expf(bp[3] * SIZE_VAR) * ph;
    const float x0 = cx - w * 0.5f, y0 = cy - h * 0.5f;
    const float x1 = cx + w * 0.5f, y1 = cy + h * 0.5f;
    bx0[k] = x0; by0[k] = y0; bx1[k] = x1; by1[k] = y1;
    area[k] = fmaxf(x1 - x0, 0.0f) * fmaxf(y1 - y0, 0.0f);
    vals[k] = v;
    keepS[k] = (v > CONF_T) ? 1 : 0;

    float* ob = out_boxes + ((long long)bc * Kc + k) * 4;
    ob[0] = x0; ob[1] = y0; ob[2] = x1; ob[3] = y1;
  }
  __syncthreads();

  for (int i = 0; i < Kc; ++i) {
    if (keepS[i] && k > i && k < Kc) {
      const float lx = fmaxf(bx0[i], bx0[k]);
      const float ly = fmaxf(by0[i], by0[k]);
      const float rx = fminf(bx1[i], bx1[k]);
      const float ry = fminf(by1[i], by1[k]);
      const float iw = fmaxf(rx - lx, 0.0f);
      const float ih = fmaxf(ry - ly, 0.0f);
      const float inter = iw * ih;
      const float uni = area[i] + area[k] - inter;
      const float iou = inter / fmaxf(uni, 1e-9f);
      if (iou > IOU_T) keepS[k] = 0;
    }
    __syncthreads();
  }

  if (k < Kc) {
    const float kv = keepS[k] ? 1.0f : 0.0f;
    out_scores[(long long)bc * Kc + k] = vals[k] * kv;
    out_keep  [(long long)bc * Kc + k] = kv;
  }
}

// -------------------------------------------------------------------------
// Host launcher
// d_in[0] = cls_logits (B*N*C f32), d_in[1] = bbox_pred (B*N*4 f32),
// d_in[2] = priors (N*4 f32).
// d_out = cand_boxes(256000) | out_scores(64000) | keep(64000), f32.
// -------------------------------------------------------------------------
extern "C" void kernel_launch(void* const* d_in, const int* in_sizes, int n_in,
                              void* d_out, int out_size, void* d_ws, size_t ws_size,
                              hipStream_t stream) {
  const float* cls_logits = (const float*)d_in[0];
  const float* bbox_pred  = (const float*)d_in[1];
  const float* priors     = (const float*)d_in[2];

  float* out = (float*)d_out;
  float* out_boxes  = out;                           // B*20*K*4 = 256000
  float* out_scores = out + (long long)Bc * NCLS * Kc * 4;      // 64000
  float* out_keep   = out_scores + (long long)Bc * NCLS * Kc;   // 64000

  // Workspace layout (floats): transposed scores | top vals | top idx
  float* ws         = (float*)d_ws;
  float* ws_scores  = ws;                                        // 20,971,520 f
  float* ws_topv    = ws + (long long)Bc * NCLS * Nc;            // 64,000 f
  int*   ws_topi    = (int*)(ws_topv + (long long)Bc * NCLS * Kc);

  // Kernel 2 needs 139264 B of dynamic LDS (> 64 KB default cap).
  static const size_t kTopkLds = (32768 + 4 * 512) * sizeof(float);
  (void)hipFuncSetAttribute((const void*)topk_kernel,
                            hipFuncAttributeMaxDynamicSharedMemorySize,
                            (int)kTopkLds);

  // 1) softmax + transpose: 1M rows / 256 per block
  softmax_transpose_kernel<<<(Bc * Nc) / 256, 256, 0, stream>>>(cls_logits, ws_scores);

  // 2) exact top-100 per (b, class)
  topk_kernel<<<Bc * NCLS, 512, kTopkLds, stream>>>(ws_scores, ws_topv, ws_topi);

  // 3) gather + decode + NMS + outputs
  nms_kernel<<<Bc * NCLS, 128, 0, stream>>>(bbox_pred, priors, ws_topv, ws_topi,
                                            out_boxes, out_scores, out_keep);
}